// SparseAutoencoder_82884278878772
// MI455X (gfx1250) — compile-verified
//
#include <hip/hip_runtime.h>
#include <hip/hip_bf16.h>

typedef float v2f __attribute__((ext_vector_type(2)));
typedef float v8f __attribute__((ext_vector_type(8)));

#define B_DIM 512
#define D_DIM 1024
#define M_DIM 32768
#define K_TOP 32
#define K_MULTI 128
#define K_AUX 64
#define NEG_INF (-3.402823466e+38f)

// ---------------------------------------------------------------- xc = x - input_bias
__global__ __launch_bounds__(256) void sae_xc_kernel(const float* __restrict__ x,
                                                     const float* __restrict__ ibias,
                                                     float* __restrict__ xc) {
    int i = blockIdx.x * 256 + threadIdx.x;           // 512*1024 total
    xc[i] = x[i] - ibias[i & (D_DIM - 1)];
}

// ---------------------------------------------------------------- W_dec [D][M] -> W_decT [M][D]
__global__ __launch_bounds__(256) void sae_transpose_kernel(const float* __restrict__ Wdec,
                                                            float* __restrict__ WdT) {
    __shared__ float tile[32][33];
    int tx = threadIdx.x & 31;
    int ty = threadIdx.x >> 5;                        // 0..7
    int mBase = blockIdx.x * 32;                      // over M (1024 blocks)
    int dBase = blockIdx.y * 32;                      // over D (32 blocks)
#pragma unroll
    for (int j = 0; j < 4; ++j) {
        int d = dBase + ty + j * 8;
        tile[ty + j * 8][tx] = Wdec[(size_t)d * M_DIM + mBase + tx];
    }
    __syncthreads();
#pragma unroll
    for (int j = 0; j < 4; ++j) {
        int m = mBase + ty + j * 8;
        WdT[(size_t)m * D_DIM + dBase + tx] = tile[tx][ty + j * 8];
    }
}

// ---------------------------------------------------------------- encoder GEMM (fp32 WMMA)
// pre[i][j] = sum_d xc[i][d] * Wenc[j][d] + nbias[j]
// wave tile: 32(i) x 64(j) via V_WMMA_F32_16X16X4_F32; block = 8 waves = 64x256 tile
__global__ __launch_bounds__(256) void sae_enc_gemm_kernel(const float* __restrict__ xc,
                                                           const float* __restrict__ Wenc,
                                                           const float* __restrict__ nbias,
                                                           float* __restrict__ pre) {
    const int wave = threadIdx.x >> 5;                // 0..7
    const int lane = threadIdx.x & 31;
    const int l16  = lane & 15;
    const int half = lane >> 4;                       // 0: K={0,1}, 1: K={2,3}
    const int wi = wave >> 2;                         // 0..1 (i dir)
    const int wj = wave & 3;                          // 0..3 (j dir)
    const int i_base = blockIdx.y * 64 + wi * 32;
    const int j_base = blockIdx.x * 256 + wj * 64;

    v8f acc[2][4] = {};

    // A-matrix 16x4 fp32 layout: lanes 0-15 rows 0-15, VGPR0=K(2*half), VGPR1=K(2*half+1)
    const float* arow0 = xc + (size_t)(i_base + l16) * D_DIM;
    const float* arow1 = xc + (size_t)(i_base + 16 + l16) * D_DIM;
    // B-matrix 4x16 fp32: lanes 0-15 cols 0-15; rows K(2*half), K(2*half+1).
    // B[k][n] = Wenc[col_n][k] -> contiguous along Wenc rows.
    const float* brow[4];
#pragma unroll
    for (int jj = 0; jj < 4; ++jj)
        brow[jj] = Wenc + (size_t)(j_base + jj * 16 + l16) * D_DIM;

    for (int k = 0; k < D_DIM; k += 4) {
        const int kk = k + half * 2;
        v2f a0 = *(const v2f*)(arow0 + kk);
        v2f a1 = *(const v2f*)(arow1 + kk);
#pragma unroll
        for (int jj = 0; jj < 4; ++jj) {
            v2f bf = *(const v2f*)(brow[jj] + kk);
            acc[0][jj] = __builtin_amdgcn_wmma_f32_16x16x4_f32(
                false, a0, false, bf, (short)0, acc[0][jj], false, false);
            acc[1][jj] = __builtin_amdgcn_wmma_f32_16x16x4_f32(
                false, a1, false, bf, (short)0, acc[1][jj], false, false);
        }
    }

    // C/D layout: lane N = l16; VGPR v -> M = v + 8*half
#pragma unroll
    for (int ii = 0; ii < 2; ++ii) {
#pragma unroll
        for (int jj = 0; jj < 4; ++jj) {
            int col = j_base + jj * 16 + l16;
            float nb = nbias[col];
#pragma unroll
            for (int v = 0; v < 8; ++v) {
                int rowi = i_base + ii * 16 + half * 8 + v;
                pre[(size_t)rowi * M_DIM + col] = acc[ii][jj][v] + nb;
            }
        }
    }
}

// ---------------------------------------------------------------- per-row top-128 (multik; top-32 prefix)
__global__ __launch_bounds__(256) void sae_topk_kernel(const float* __restrict__ pre,
                                                       float* __restrict__ mkv,
                                                       int* __restrict__ mki,
                                                       float* __restrict__ o_ti,
                                                       float* __restrict__ o_tv) {
    extern __shared__ float smem[];                   // M_DIM floats (128 KB of 320 KB LDS)
    __shared__ float rv[256];
    __shared__ int   ri[256];
    const int row = blockIdx.x, tid = threadIdx.x;
    const float* p = pre + (size_t)row * M_DIM;
    for (int i = tid; i < M_DIM; i += 256) smem[i] = p[i];
    __syncthreads();

    for (int t = 0; t < K_MULTI; ++t) {
        float bv = NEG_INF; int bi = 0x7fffffff;
        for (int i = tid; i < M_DIM; i += 256) {      // strict '>' keeps lowest index on ties
            float v = smem[i];
            if (v > bv) { bv = v; bi = i; }
        }
        rv[tid] = bv; ri[tid] = bi;
        __syncthreads();
        for (int s = 128; s > 0; s >>= 1) {
            if (tid < s) {
                float ov = rv[tid + s]; int oi = ri[tid + s];
                if (ov > rv[tid] || (ov == rv[tid] && oi < ri[tid])) { rv[tid] = ov; ri[tid] = oi; }
            }
            __syncthreads();
        }
        if (tid == 0) {
            int wi = ri[0]; float wv = rv[0];
            float r = wv > 0.f ? wv : 0.f;            // relu after selection
            mki[row * K_MULTI + t] = wi;
            mkv[row * K_MULTI + t] = r;
            if (t < K_TOP) {
                o_ti[row * K_TOP + t] = (float)wi;
                o_tv[row * K_TOP + t] = r;
            }
            smem[wi] = NEG_INF;
        }
        __syncthreads();
    }
}

// ---------------------------------------------------------------- per-row aux top-64 over dead-masked pre_act
__global__ __launch_bounds__(256) void sae_aux_kernel(const float* __restrict__ pre,
                                                      const int* __restrict__ steps,
                                                      float* __restrict__ o_ai,
                                                      float* __restrict__ o_av) {
    extern __shared__ float smem[];
    __shared__ float rv[256];
    __shared__ int   ri[256];
    const int row = blockIdx.x, tid = threadIdx.x;
    const float* p = pre + (size_t)row * M_DIM;
    for (int i = tid; i < M_DIM; i += 256) {
        bool dead = (steps[i] + 1) > 256;             // DEAD_THRESH
        smem[i] = dead ? p[i] : 0.0f;                 // pre_act * mask
    }
    __syncthreads();

    for (int t = 0; t < K_AUX; ++t) {
        float bv = NEG_INF; int bi = 0x7fffffff;
        for (int i = tid; i < M_DIM; i += 256) {
            float v = smem[i];
            if (v > bv) { bv = v; bi = i; }
        }
        rv[tid] = bv; ri[tid] = bi;
        __syncthreads();
        for (int s = 128; s > 0; s >>= 1) {
            if (tid < s) {
                float ov = rv[tid + s]; int oi = ri[tid + s];
                if (ov > rv[tid] || (ov == rv[tid] && oi < ri[tid])) { rv[tid] = ov; ri[tid] = oi; }
            }
            __syncthreads();
        }
        if (tid == 0) {
            int wi = ri[0]; float wv = rv[0];
            o_ai[row * K_AUX + t] = (float)wi;
            o_av[row * K_AUX + t] = wv > 0.f ? wv : 0.f;
            smem[wi] = NEG_INF;
        }
        __syncthreads();
    }
}

// ---------------------------------------------------------------- sparse decode: 4 reconstructions in one pass
__global__ __launch_bounds__(256) void sae_decode_kernel(const float* __restrict__ WdT,
                                                         const float* __restrict__ mkv,
                                                         const int* __restrict__ mki,
                                                         const float* __restrict__ ibias,
                                                         float* __restrict__ o_rec,
                                                         float* __restrict__ o_mk,
                                                         float* __restrict__ o_m0,
                                                         float* __restrict__ o_m1) {
    __shared__ float sv[K_MULTI];
    __shared__ int   si[K_MULTI];
    const int row = blockIdx.x, tid = threadIdx.x;
    if (tid < K_MULTI) { sv[tid] = mkv[row * K_MULTI + tid]; si[tid] = mki[row * K_MULTI + tid]; }
    __syncthreads();
    for (int d = tid; d < D_DIM; d += 256) {
        float r_top = 0.f, r_mk = 0.f, r_m0 = 0.f, r_m1 = 0.f;
#pragma unroll 4
        for (int t = 0; t < K_MULTI; ++t) {
            float w = WdT[(size_t)si[t] * D_DIM + d]; // coalesced: contiguous in d
            float c = sv[t] * w;
            r_mk += c;
            if (t < K_TOP) {
                r_top += c;
                if (si[t] < 16384) { r_m1 += c; if (si[t] < 4096) r_m0 += c; }
            }
        }
        float ib = ibias[d];
        size_t o = (size_t)row * D_DIM + d;
        o_rec[o] = r_top + ib;
        o_mk[o]  = r_mk + ib;
        o_m0[o]  = r_m0 + ib;
        o_m1[o]  = r_m1 + ib;
    }
}

// ---------------------------------------------------------------- launch
extern "C" void kernel_launch(void* const* d_in, const int* in_sizes, int n_in,
                              void* d_out, int out_size, void* d_ws, size_t ws_size,
                              hipStream_t stream) {
    const float* x     = (const float*)d_in[0];   // [512,1024]
    const float* Wenc  = (const float*)d_in[1];   // [32768,1024]
    const float* Wdec  = (const float*)d_in[2];   // [1024,32768]
    const float* ibias = (const float*)d_in[3];   // [1024]
    const float* nbias = (const float*)d_in[4];   // [32768]
    const int*   steps = (const int*)d_in[5];     // [32768]

    float* out = (float*)d_out;
    // output tuple layout (flat, return order)
    const size_t BD = (size_t)B_DIM * D_DIM;
    float* o_rec = out;                                 // reconstruction
    float* o_mk  = out + BD;                            // multik reconstruction
    float* o_m0  = out + 2 * BD;                        // matryoshka 4096
    float* o_m1  = out + 3 * BD;                        // matryoshka 16384
    float* o_ti  = out + 4 * BD;                        // topk indices (as float) [512,32]
    float* o_tv  = o_ti + (size_t)B_DIM * K_TOP;        // topk values
    float* o_ai  = o_tv + (size_t)B_DIM * K_TOP;        // aux indices  [512,64]
    float* o_av  = o_ai + (size_t)B_DIM * K_AUX;        // aux values

    // workspace layout
    float* ws_pre = (float*)d_ws;                       // [512,32768]  64 MB
    float* ws_xc  = ws_pre + (size_t)B_DIM * M_DIM;     // [512,1024]    2 MB
    float* ws_wdT = ws_xc + (size_t)B_DIM * D_DIM;      // [32768,1024] 128 MB
    float* ws_mkv = ws_wdT + (size_t)M_DIM * D_DIM;     // [512,128]
    int*   ws_mki = (int*)(ws_mkv + (size_t)B_DIM * K_MULTI);

    sae_xc_kernel<<<(B_DIM * D_DIM) / 256, 256, 0, stream>>>(x, ibias, ws_xc);
    sae_transpose_kernel<<<dim3(M_DIM / 32, D_DIM / 32), 256, 0, stream>>>(Wdec, ws_wdT);
    sae_enc_gemm_kernel<<<dim3(M_DIM / 256, B_DIM / 64), 256, 0, stream>>>(ws_xc, Wenc, nbias, ws_pre);
    sae_topk_kernel<<<B_DIM, 256, M_DIM * sizeof(float), stream>>>(ws_pre, ws_mkv, ws_mki, o_ti, o_tv);
    sae_aux_kernel<<<B_DIM, 256, M_DIM * sizeof(float), stream>>>(ws_pre, steps, o_ai, o_av);
    sae_decode_kernel<<<B_DIM, 256, 0, stream>>>(ws_wdT, ws_mkv, ws_mki, ibias, o_rec, o_mk, o_m0, o_m1);
}